// Model_45990509805946
// MI455X (gfx1250) — compile-verified
//
#include <hip/hip_runtime.h>

// CDNA5 / gfx1250: memory-bound stencil kernel.
// 314 MB unique traffic -> ~13.5 us floor @ 23.3 TB/s; ~0.5 GFLOP total, so no
// WMMA (K=3 dot per pixel). This revision removes the integer div/mod index
// math via a 3D launch and replaces IEEE divides with v_rcp_f32-based fast
// safe_div to keep VALU well under the HBM roofline.

typedef float v4f __attribute__((ext_vector_type(4)));
typedef float v2f __attribute__((ext_vector_type(2)));

namespace {
constexpr int   kH = 480;
constexpr int   kW = 640;
constexpr int   kB = 32;
constexpr float kCX = 313.04f;
constexpr float kCY = 238.44f;
constexpr float kRFX = 1.0f / 582.62f;
constexpr float kRFY = 1.0f / 582.69f;
constexpr int   kSegs = kW / 4;       // 160 float4 segments per row = 5 wave32
constexpr int   kRowsPerBlock = 2;    // blockDim = (160, 2) = 10 waves
}

// safe_div(a,b) = (b==0) ? 0 : a/b, with the divide done as a * v_rcp_f32(b).
// v_rcp_f32 is a TRANS op (co-issues with VALU), ~1 ulp — fine here, and it
// turns an ~8-instruction IEEE div chain into cmp+rcp+mul+cndmask.
__device__ __forceinline__ float safe_div(float a, float b) {
  float q = a * __builtin_amdgcn_rcpf(b);
  return (b == 0.0f) ? 0.0f : q;
}

extern "C" __global__ __launch_bounds__(kSegs * kRowsPerBlock)
void depth_prop_kernel(const float* __restrict__ depth,
                       const float* __restrict__ normal,
                       float* __restrict__ out) {
  const int x0 = threadIdx.x << 2;                         // 16B-aligned column
  const int y  = blockIdx.y * kRowsPerBlock + threadIdx.y; // 0..479
  const int b  = blockIdx.z;                               // 0..31

  const size_t plane = (size_t)kH * kW;
  const float* d0 = depth + (size_t)b * plane;
  const float* n0 = normal + (size_t)b * 3 * plane;   // nx
  const float* n1 = n0 + plane;                       // ny
  const float* n2 = n1 + plane;                       // nz

  // gfx1250 speculative prefetch of a depth row the y+2 tap will want soon.
  if (y + 4 < kH) __builtin_prefetch(d0 + (size_t)(y + 4) * kW + x0, 0, 1);

  const size_t ry = (size_t)y * kW;
  const int ym1 = (y > 0) ? y - 1 : 0;
  const int yp1 = (y < kH - 1) ? y + 1 : kH - 1;

  // ---- loads: rows y-1 / y+1 normals (b128 each) ----
  v4f nxm = *(const v4f*)(n0 + (size_t)ym1 * kW + x0);
  v4f nym = *(const v4f*)(n1 + (size_t)ym1 * kW + x0);
  v4f nzm = *(const v4f*)(n2 + (size_t)ym1 * kW + x0);
  v4f nxp = *(const v4f*)(n0 + (size_t)yp1 * kW + x0);
  v4f nyp = *(const v4f*)(n1 + (size_t)yp1 * kW + x0);
  v4f nzp = *(const v4f*)(n2 + (size_t)yp1 * kW + x0);

  // ---- row-y normals, columns x0-1 .. x0+4 (b128 + 2 scalars / channel) ----
  float nrx[6], nry[6], nrz[6];
  {
    const int cm = (x0 > 0) ? x0 - 1 : 0;          // used only when masked off
    const int cp = (x0 + 4 < kW) ? x0 + 4 : kW - 1;
    v4f a = *(const v4f*)(n0 + ry + x0);
    v4f c = *(const v4f*)(n1 + ry + x0);
    v4f e = *(const v4f*)(n2 + ry + x0);
    nrx[0] = n0[ry + cm]; nrx[1] = a[0]; nrx[2] = a[1]; nrx[3] = a[2]; nrx[4] = a[3]; nrx[5] = n0[ry + cp];
    nry[0] = n1[ry + cm]; nry[1] = c[0]; nry[2] = c[1]; nry[3] = c[2]; nry[4] = c[3]; nry[5] = n1[ry + cp];
    nrz[0] = n2[ry + cm]; nrz[1] = e[0]; nrz[2] = e[1]; nrz[3] = e[2]; nrz[4] = e[3]; nrz[5] = n2[ry + cp];
  }

  // ---- depth row y, columns x0-2 .. x0+5 (b64 + b128 + b64), zero-filled ----
  float dr[8];
  {
    v4f dm = *(const v4f*)(d0 + ry + x0);
    dr[2] = dm[0]; dr[3] = dm[1]; dr[4] = dm[2]; dr[5] = dm[3];
    if (x0 > 0) { v2f t = *(const v2f*)(d0 + ry + x0 - 2); dr[0] = t[0]; dr[1] = t[1]; }
    else        { dr[0] = 0.0f; dr[1] = 0.0f; }
    if (x0 + 5 < kW) { v2f t = *(const v2f*)(d0 + ry + x0 + 4); dr[6] = t[0]; dr[7] = t[1]; }
    else             { dr[6] = 0.0f; dr[7] = 0.0f; }
  }

  // ---- depth rows y-2 / y+2 (b128), zero-filled at borders ----
  v4f du2 = {0.0f, 0.0f, 0.0f, 0.0f};
  v4f dd2 = {0.0f, 0.0f, 0.0f, 0.0f};
  if (y >= 2)     du2 = *(const v4f*)(d0 + (size_t)(y - 2) * kW + x0);
  if (y + 2 < kH) dd2 = *(const v4f*)(d0 + (size_t)(y + 2) * kW + x0);

  // ---- back-projection grid values ----
  float u[8];                                  // u(c) for c = x0-2 .. x0+5
#pragma unroll
  for (int k = 0; k < 8; ++k) u[k] = ((float)(x0 - 2 + k) - kCX) * kRFX;
  const float vy  = ((float)y       - kCY) * kRFY;   // v(y)
  const float vm2 = ((float)(y - 2) - kCY) * kRFY;   // v(y-2) = v((y-1)-1)
  const float vp2 = ((float)(y + 2) - kCY) * kRFY;   // v(y+2) = v((y+1)+1)

  // ABC_left/right for row-y columns c = x0-1+j, j = 0..5
  float Al[6], Ar[6];
#pragma unroll
  for (int j = 0; j < 6; ++j) {
    float t = fmaf(nry[j], vy, nrz[j]);
    Al[j] = fmaf(nrx[j], u[j],     t);   // uses u(c-1)
    Ar[j] = fmaf(nrx[j], u[j + 2], t);   // uses u(c+1)
  }

  v4f oU, oD, oL, oR;
#pragma unroll
  for (int i = 0; i < 4; ++i) {
    const int x = x0 + i;
    // horizontal: ratios at columns x-1 (j=i) and x+1 (j=i+2)
    float r = safe_div(Al[i],     Ar[i])     * dr[i];       // depth(y, x-2)
    float l = safe_div(Ar[i + 2], Al[i + 2]) * dr[i + 4];   // depth(y, x+2)
    // vertical: ratios at rows y-1 / y+1, column x (u index i+2)
    const float um = u[i + 2];
    float tuu = fmaf(nym[i], vm2, nzm[i]);                  // ABC_up(y-1) tail
    float tud = fmaf(nym[i], vy,  nzm[i]);                  // ABC_down(y-1) tail
    float d   = safe_div(fmaf(nxm[i], um, tuu), fmaf(nxm[i], um, tud)) * du2[i];
    float tdd = fmaf(nyp[i], vp2, nzp[i]);                  // ABC_down(y+1) tail
    float tdu = fmaf(nyp[i], vy,  nzp[i]);                  // ABC_up(y+1) tail
    float up  = safe_div(fmaf(nxp[i], um, tdd), fmaf(nxp[i], um, tdu)) * dd2[i];

    oR[i] = (x >= 1)      ? r  : 0.0f;
    oL[i] = (x < kW - 1)  ? l  : 0.0f;
    oD[i] = (y >= 1)      ? d  : 0.0f;
    oU[i] = (y < kH - 1)  ? up : 0.0f;
  }

  // ---- streaming stores: non-temporal b128 (output never re-read) ----
  float* ob = out + (size_t)b * 4 * plane + ry + x0;
  __builtin_nontemporal_store(oU, (v4f*)(ob));               // ch 0: z_o_up
  __builtin_nontemporal_store(oD, (v4f*)(ob + plane));       // ch 1: z_o_down
  __builtin_nontemporal_store(oL, (v4f*)(ob + 2 * plane));   // ch 2: z_o_left
  __builtin_nontemporal_store(oR, (v4f*)(ob + 3 * plane));   // ch 3: z_o_right
}

extern "C" void kernel_launch(void* const* d_in, const int* in_sizes, int n_in,
                              void* d_out, int out_size, void* d_ws, size_t ws_size,
                              hipStream_t stream) {
  (void)in_sizes; (void)n_in; (void)out_size; (void)d_ws; (void)ws_size;
  const float* depth  = (const float*)d_in[0];   // [B,1,H,W] f32
  const float* normal = (const float*)d_in[1];   // [B,3,H,W] f32
  float* out = (float*)d_out;                    // [B,4,H,W] f32
  dim3 block(kSegs, kRowsPerBlock, 1);           // (160, 2) = 10 wave32
  dim3 grid(1, kH / kRowsPerBlock, kB);          // (1, 240, 32)
  depth_prop_kernel<<<grid, block, 0, stream>>>(depth, normal, out);
}